// ForecastingAE_65790309040712
// MI455X (gfx1250) — compile-verified
//
#include <hip/hip_runtime.h>

// Soft guard: on the host pass the AMDGCN builtin doesn't exist, so the WMMA
// body compiles out (host stub only needs to parse). The device (gfx1250)
// pass has the builtin and compiles the real kernel.
#if __has_builtin(__builtin_amdgcn_wmma_f32_16x16x4_f32)
#define HAS_F32_WMMA 1
#endif

typedef __attribute__((ext_vector_type(2))) float v2f;
typedef __attribute__((ext_vector_type(8))) float v8f;

#define BB 1024
#define MM 4
#define LL 100
#define DD 512
#define HH 512
#define DEC 128

// ---------------------------------------------------------------------------
// Kernel 1: masked average pooling of embedding gathers.
// One block per (batch, segment). 256 threads; thread t owns columns t, t+256.
// Segments 3,2,0 -> xs[0..2] (scaled by masks), segment 1 -> x_future output.
// ---------------------------------------------------------------------------
__global__ __launch_bounds__(256) void pool_kernel(
    const int* __restrict__ inputs, const float* __restrict__ masks,
    const float* __restrict__ table, float* __restrict__ xs,
    float* __restrict__ x_future) {
  const int b = blockIdx.x >> 2;
  const int seg = blockIdx.x & 3;
  const int tid = threadIdx.x;

  __shared__ int s_tok[LL];
  const int* tok = inputs + ((size_t)b * MM + seg) * LL;
  if (tid < LL) s_tok[tid] = tok[tid];
  __syncthreads();

  float acc0 = 0.f, acc1 = 0.f;
  int cnt = 0;
  for (int l = 0; l < LL; ++l) {
    const int t = s_tok[l];
    if (t != 0) {
      ++cnt;
      const float* row = table + (size_t)t * DD;
      acc0 += row[tid];
      acc1 += row[tid + 256];
    }
  }
  const float inv = 1.0f / fmaxf((float)cnt, 1.0f);
  acc0 *= inv;
  acc1 *= inv;

  if (seg == 1) {
    float* dst = x_future + (size_t)b * DD;
    dst[tid] = acc0;
    dst[tid + 256] = acc1;
  } else {
    // past = [3, 2, 0] -> timestep slot 0, 1, 2
    const int slot = (seg == 3) ? 0 : (seg == 2) ? 1 : 2;
    const float m = masks[b * MM + seg];
    float* dst = xs + ((size_t)slot * BB + b) * DD;
    dst[tid] = acc0 * m;
    dst[tid + 256] = acc1 * m;
  }
}

// ---------------------------------------------------------------------------
// Kernel 2: dual-input GEMM with WMMA f32 16x16x4 (exact f32 math).
//   out[M,N] = act( A1[M,K1] @ W1[N,K1]^T  (+ A2[M,K2] @ W2[N,K2]^T)
//                   + bias1[N] (+ bias2[N]) )
// Block = 128 threads = 4 waves. Wave w computes a 16x64 strip:
// rows [by*64 + 16w, +16), cols [bx*64, +64) via 4 independent accumulators.
// Per k-step: all 5 fragment loads (a, b0..b3) are issued BEFORE the 4-WMMA
// burst so the first WMMA waits on loadcnt<=3 and the remaining loads overlap
// matrix-pipe execution (instead of load->wait0->wmma serialization).
//
// Fragment layout (ISA 7.12.2, 32-bit 16x4 A / 4x16 B): per lane a float2 at
//   row = tileBase + (lane & 15),  k = kBase + (lane >> 4) * 2
// which is identical for A (row-major M,K) and B = W^T (W row-major N,K).
// ---------------------------------------------------------------------------
__device__ inline v2f load_frag(const float* __restrict__ p, int base_row,
                                int k, int ld, int lane) {
  const float* q =
      p + (size_t)(base_row + (lane & 15)) * ld + k + ((lane >> 4) << 1);
  v2f r = {q[0], q[1]};
  return r;
}

__global__ __launch_bounds__(128) void gemm16_dual(
    const float* __restrict__ A1, const float* __restrict__ W1, int K1,
    const float* __restrict__ A2, const float* __restrict__ W2, int K2,
    const float* __restrict__ bias1, const float* __restrict__ bias2,
    float* __restrict__ C, int N, int relu) {
#ifdef HAS_F32_WMMA
  const int lane = threadIdx.x & 31;
  const int wave = threadIdx.x >> 5;
  const int tileM = blockIdx.y * 64 + wave * 16;
  const int tileN = blockIdx.x * 64;

  v8f acc[4];
#pragma unroll
  for (int t = 0; t < 4; ++t)
    acc[t] = v8f{0.f, 0.f, 0.f, 0.f, 0.f, 0.f, 0.f, 0.f};

  for (int k = 0; k < K1; k += 4) {
    // batch all loads of this k-step first
    v2f a = load_frag(A1, tileM, k, K1, lane);
    v2f b[4];
#pragma unroll
    for (int t = 0; t < 4; ++t) b[t] = load_frag(W1, tileN + 16 * t, k, K1, lane);
    // then the WMMA burst (4 independent accumulator chains)
#pragma unroll
    for (int t = 0; t < 4; ++t)
      acc[t] = __builtin_amdgcn_wmma_f32_16x16x4_f32(false, a, false, b[t],
                                                     (short)0, acc[t], false,
                                                     false);
  }
  if (A2 != nullptr) {  // uniform branch: EXEC stays all-ones
    for (int k = 0; k < K2; k += 4) {
      v2f a = load_frag(A2, tileM, k, K2, lane);
      v2f b[4];
#pragma unroll
      for (int t = 0; t < 4; ++t)
        b[t] = load_frag(W2, tileN + 16 * t, k, K2, lane);
#pragma unroll
      for (int t = 0; t < 4; ++t)
        acc[t] = __builtin_amdgcn_wmma_f32_16x16x4_f32(false, a, false, b[t],
                                                       (short)0, acc[t], false,
                                                       false);
    }
  }

  const int rowBase = tileM + ((lane >> 4) << 3);
#pragma unroll
  for (int t = 0; t < 4; ++t) {
    const int col = tileN + 16 * t + (lane & 15);
    float bias = bias1[col];
    if (bias2 != nullptr) bias += bias2[col];
#pragma unroll
    for (int j = 0; j < 8; ++j) {
      float v = acc[t][j] + bias;
      if (relu) v = fmaxf(v, 0.f);
      C[(size_t)(rowBase + j) * N + col] = v;
    }
  }
#endif  // HAS_F32_WMMA
}

// ---------------------------------------------------------------------------
// Host-side orchestration. 1 pool + 6 RNN GEMMs + 2 decoder GEMMs, all on
// `stream`; kernel boundaries provide the inter-timestep global sync.
// ---------------------------------------------------------------------------
extern "C" void kernel_launch(void* const* d_in, const int* in_sizes, int n_in,
                              void* d_out, int out_size, void* d_ws,
                              size_t ws_size, hipStream_t stream) {
  const int* inputs = (const int*)d_in[0];
  const float* masks = (const float*)d_in[1];
  const float* table = (const float*)d_in[2];
  const float* w_ih0 = (const float*)d_in[3];
  const float* w_hh0 = (const float*)d_in[4];
  const float* b_ih0 = (const float*)d_in[5];
  const float* b_hh0 = (const float*)d_in[6];
  const float* w_ih1 = (const float*)d_in[7];
  const float* w_hh1 = (const float*)d_in[8];
  const float* b_ih1 = (const float*)d_in[9];
  const float* b_hh1 = (const float*)d_in[10];
  const float* w_d1 = (const float*)d_in[11];
  const float* b_d1 = (const float*)d_in[12];
  const float* w_d2 = (const float*)d_in[13];
  const float* b_d2 = (const float*)d_in[14];

  float* out = (float*)d_out;
  float* x_future_hat = out;                  // [B, D]
  float* x_future = out + (size_t)BB * DD;    // [B, D]
  float* last_h = out + 2 * (size_t)BB * DD;  // [B, H]

  char* ws = (char*)d_ws;
  float* xs = (float*)ws;                                 // 3*B*D = 6 MB
  float* h0a = (float*)(ws + (size_t)6 * 1024 * 1024);    // 2 MB
  float* h0b = (float*)(ws + (size_t)8 * 1024 * 1024);    // 2 MB
  float* h1a = (float*)(ws + (size_t)10 * 1024 * 1024);   // 2 MB
  float* h1b = (float*)(ws + (size_t)12 * 1024 * 1024);   // 2 MB
  float* hdec = (float*)(ws + (size_t)14 * 1024 * 1024);  // 0.5 MB

  // 1) pooled, masked embeddings
  pool_kernel<<<dim3(BB * MM), 256, 0, stream>>>(inputs, masks, table, xs,
                                                 x_future);

  const dim3 blk(128);
  const dim3 gH(HH / 64, BB / 64);      // N=512: 8 x 16 blocks
  const dim3 gDec1(DEC / 64, BB / 64);  // N=128: 2 x 16 blocks
  const float* xs0 = xs;
  const float* xs1 = xs + (size_t)BB * DD;
  const float* xs2 = xs + 2 * (size_t)BB * DD;

  // t = 0 (h == 0 -> skip recurrent matmul, biases still applied)
  gemm16_dual<<<gH, blk, 0, stream>>>(xs0, w_ih0, DD, nullptr, nullptr, 0,
                                      b_ih0, b_hh0, h0a, HH, 1);
  gemm16_dual<<<gH, blk, 0, stream>>>(h0a, w_ih1, HH, nullptr, nullptr, 0,
                                      b_ih1, b_hh1, h1a, HH, 1);
  // t = 1
  gemm16_dual<<<gH, blk, 0, stream>>>(xs1, w_ih0, DD, h0a, w_hh0, HH, b_ih0,
                                      b_hh0, h0b, HH, 1);
  gemm16_dual<<<gH, blk, 0, stream>>>(h0b, w_ih1, HH, h1a, w_hh1, HH, b_ih1,
                                      b_hh1, h1b, HH, 1);
  // t = 2 (final h1 written straight to the last_h output region)
  gemm16_dual<<<gH, blk, 0, stream>>>(xs2, w_ih0, DD, h0b, w_hh0, HH, b_ih0,
                                      b_hh0, h0a, HH, 1);
  gemm16_dual<<<gH, blk, 0, stream>>>(h0a, w_ih1, HH, h1b, w_hh1, HH, b_ih1,
                                      b_hh1, last_h, HH, 1);
  // decoder
  gemm16_dual<<<gDec1, blk, 0, stream>>>(last_h, w_d1, HH, nullptr, nullptr, 0,
                                         b_d1, nullptr, hdec, DEC, 1);
  gemm16_dual<<<gH, blk, 0, stream>>>(hdec, w_d2, DEC, nullptr, nullptr, 0,
                                      b_d2, nullptr, x_future_hat, DD, 0);
}